// MacroGCN_20203526160738
// MI455X (gfx1250) — compile-verified
//
#include <hip/hip_runtime.h>
#include <hip/hip_bf16.h>

typedef __attribute__((ext_vector_type(2))) float v2f;
typedef __attribute__((ext_vector_type(8))) float v8f;

// ---------------- degree / normalization ----------------

__global__ void k_init_deg(float* __restrict__ deg, int n) {
    int i = blockIdx.x * blockDim.x + threadIdx.x;
    if (i < n) deg[i] = 1.0f;  // self-loop contributes 1
}

__global__ void k_edge_deg(const int* __restrict__ dst, float* __restrict__ deg, int nE) {
    int e = blockIdx.x * blockDim.x + threadIdx.x;
    if (e < nE) atomicAdd(&deg[dst[e]], 1.0f);
}

__global__ void k_rsqrt_inplace(float* __restrict__ deg, int n) {
    int i = blockIdx.x * blockDim.x + threadIdx.x;
    if (i < n) deg[i] = rsqrtf(deg[i]);
}

// ---- fp32 WMMA GEMM, 32x16 tile per wave, fused self-loop epilogue ----
// H[M,N] = A[M,K] @ B[K,N];  AGG[m,n] = H[m,n] * dinv[m]^2  (self-loop seed
// for the subsequent edge scatter).  One wave computes two stacked 16x16
// subtiles sharing the same B fragment -> 2 v_wmma per B fetch.
// grid = (M/32, N/16), block = 32.  Requires M%32==0, N%16==0, K%4==0
// (true here: M=100000, N in {128,32}, K=128).

__global__ void k_gemm_wmma_fused(const float* __restrict__ A, const float* __restrict__ B,
                                  const float* __restrict__ dinv,
                                  float* __restrict__ H, float* __restrict__ AGG,
                                  int K, int N) {
    const int lane = threadIdx.x & 31;
    const int mt   = blockIdx.x;                   // 32-row tile index
    const int nt   = blockIdx.y;                   // 16-col tile index

    const int nc = (nt << 4) + (lane & 15);        // B/C column this lane supplies
    const int kq = (lane >> 4) << 1;               // 0 (lanes 0-15) or 2 (lanes 16-31)

    const int r0 = (mt << 5) + (lane & 15);        // A row, subtile 0
    const float* __restrict__ a0p = A + (size_t)r0 * K + kq;
    const float* __restrict__ a1p = a0p + (size_t)16 * K;   // subtile 1: rows +16

    v8f c0 = {}, c1 = {};
    for (int k = 0; k < K; k += 4) {
        v2f a0, a1, b;
        a0.x = a0p[k];     a0.y = a0p[k + 1];
        a1.x = a1p[k];     a1.y = a1p[k + 1];
        b.x  = B[(size_t)(k + kq) * N + nc];
        b.y  = B[(size_t)(k + kq + 1) * N + nc];
        c0 = __builtin_amdgcn_wmma_f32_16x16x4_f32(false, a0, false, b, (short)0, c0,
                                                   false, false);
        c1 = __builtin_amdgcn_wmma_f32_16x16x4_f32(false, a1, false, b, (short)0, c1,
                                                   false, false);
    }

    // C/D layout: VGPR r holds row M = r + 8*(lane>>4), col N = lane&15
    const int m0 = (mt << 5) + ((lane >> 4) << 3);
#pragma unroll
    for (int r = 0; r < 8; ++r) {
        const int ma = m0 + r;          // subtile 0 row
        const int mb = ma + 16;         // subtile 1 row
        float wa = dinv[ma]; wa *= wa;
        float wb = dinv[mb]; wb *= wb;
        H[(size_t)ma * N + nc]   = c0[r];
        H[(size_t)mb * N + nc]   = c1[r];
        AGG[(size_t)ma * N + nc] = c0[r] * wa;
        AGG[(size_t)mb * N + nc] = c1[r] * wb;
    }
}

// ---------------- edge scatter: agg[dst] += h[src] * dinv[src]*dinv[dst] ----------------
// Thread handles (edge, 4 consecutive features); for D=128 a wave32 = one full edge row.

__global__ void k_scatter(const float* __restrict__ h, const int* __restrict__ src,
                          const int* __restrict__ dst, const float* __restrict__ dinv,
                          float* __restrict__ agg, int nE, int D) {
    size_t tid = (size_t)blockIdx.x * blockDim.x + threadIdx.x;
    const int per = D >> 2;
    size_t total = (size_t)nE * per;
    if (tid >= total) return;
    int e  = (int)(tid / per);
    int f4 = (int)(tid % per) << 2;
    const int s = src[e];
    const int d = dst[e];
    const float nrm = dinv[s] * dinv[d];
    const float4 hv = *(const float4*)(h + (size_t)s * D + f4);
    float* ap = agg + (size_t)d * D + f4;
    atomicAdd(ap + 0, hv.x * nrm);
    atomicAdd(ap + 1, hv.y * nrm);
    atomicAdd(ap + 2, hv.z * nrm);
    atomicAdd(ap + 3, hv.w * nrm);
}

// ---------------- bias + relu (in place) ----------------

__global__ void k_bias_relu(float* __restrict__ x, const float* __restrict__ b,
                            int n, int D) {
    size_t tid = (size_t)blockIdx.x * blockDim.x + threadIdx.x;
    size_t total = (size_t)n * D;
    if (tid >= total) return;
    int f = (int)(tid % D);
    x[tid] = fmaxf(x[tid] + b[f], 0.0f);
}

// ---------------- bias + log_softmax over 32 labels (one wave32 per row) ----------------

__global__ void k_bias_logsoftmax32(float* __restrict__ out, const float* __restrict__ b,
                                    int n) {
    const int row  = blockIdx.x * (blockDim.x >> 5) + (threadIdx.x >> 5);
    const int lane = threadIdx.x & 31;
    if (row >= n) return;
    float v = out[(size_t)row * 32 + lane] + b[lane];
    float mx = v;
#pragma unroll
    for (int off = 16; off > 0; off >>= 1)
        mx = fmaxf(mx, __shfl_xor(mx, off, 32));
    float ex = __expf(v - mx);
    float s = ex;
#pragma unroll
    for (int off = 16; off > 0; off >>= 1)
        s += __shfl_xor(s, off, 32);
    out[(size_t)row * 32 + lane] = v - mx - logf(s);
}

// ---------------- launch ----------------

extern "C" void kernel_launch(void* const* d_in, const int* in_sizes, int n_in,
                              void* d_out, int out_size, void* d_ws, size_t ws_size,
                              hipStream_t stream) {
    const float* X     = (const float*)d_in[0];
    const int*   edges = (const int*)d_in[1];
    const float* W1    = (const float*)d_in[2];
    const float* b1    = (const float*)d_in[3];
    const float* W2    = (const float*)d_in[4];
    const float* b2    = (const float*)d_in[5];

    const int Dh  = in_sizes[3];               // 128
    const int Fin = in_sizes[2] / Dh;          // 128
    const int L   = in_sizes[5];               // 32
    const int N   = in_sizes[0] / Fin;         // 100000
    const int E   = in_sizes[1] / 2;           // 625000
    const int* src = edges;
    const int* dst = edges + E;

    // workspace layout: dinv | h (N*Dh) | agg (N*Dh);  h2 reuses h after layer-1 scatter
    char* ws = (char*)d_ws;
    float* dinv = (float*)ws;
    size_t off1 = ((size_t)N * 4 + 255) & ~(size_t)255;
    float* h    = (float*)(ws + off1);
    size_t off2 = off1 + (((size_t)N * Dh * 4 + 255) & ~(size_t)255);
    float* agg  = (float*)(ws + off2);
    float* h2   = h;
    float* out  = (float*)d_out;

    // --- degrees / symmetric normalization ---
    k_init_deg<<<(N + 255) / 256, 256, 0, stream>>>(dinv, N);
    k_edge_deg<<<(E + 255) / 256, 256, 0, stream>>>(dst, dinv, E);
    k_rsqrt_inplace<<<(N + 255) / 256, 256, 0, stream>>>(dinv, N);

    // --- layer 1: h = X@W1 (agg seeded with self-loop term in epilogue);
    //     agg += scatter(norm * h) ; h1 = relu(agg + b1)  (in place in agg) ---
    {
        dim3 grid(N / 32, Dh / 16);
        k_gemm_wmma_fused<<<grid, 32, 0, stream>>>(X, W1, dinv, h, agg, Fin, Dh);

        size_t stot = (size_t)E * (Dh >> 2);
        k_scatter<<<(int)((stot + 255) / 256), 256, 0, stream>>>(h, src, dst, dinv, agg, E, Dh);

        size_t btot = (size_t)N * Dh;
        k_bias_relu<<<(int)((btot + 255) / 256), 256, 0, stream>>>(agg, b1, N, Dh);
    }

    // --- layer 2: h2 = h1@W2 (out seeded with self-loop term in epilogue);
    //     out += scatter(norm * h2) ; log_softmax(out + b2) ---
    {
        dim3 grid(N / 32, L / 16);
        k_gemm_wmma_fused<<<grid, 32, 0, stream>>>(agg, W2, dinv, h2, out, Dh, L);

        size_t stot = (size_t)E * (L >> 2);
        k_scatter<<<(int)((stot + 255) / 256), 256, 0, stream>>>(h2, src, dst, dinv, out, E, L);

        k_bias_logsoftmax32<<<(N + 7) / 8, 256, 0, stream>>>(out, b2, N);
    }
}